// AvgPool2d_53042846105941
// MI455X (gfx1250) — compile-verified
//
#include <hip/hip_runtime.h>

typedef __attribute__((ext_vector_type(2))) float v2f;
typedef __attribute__((ext_vector_type(8))) float v8f;

#define IN_DIM  18496      // C * 34 * 34
#define OUT_DIM 4624       // C * 17 * 17
#define BATCH   32
#define KSPLIT  8
#define NTILES  (OUT_DIM / 16)   // 289
#define WAVES_PER_BLOCK 4
#define RSTEP   (8 * IN_DIM)     // v2f-element stride from row m to row m+16

// One wave computes a 32(M) x 16(N) f32 partial-product tile over one K-chunk
// using V_WMMA_F32_16X16X4_F32. Mask is folded into the A-fragment load.
// Weight (streamed once from HBM) is loaded non-temporally to keep x/mask
// resident in L2 across all 289 N-tiles.
__global__ __launch_bounds__(WAVES_PER_BLOCK * 32)
void avgpool_toeplitz_wmma(const float* __restrict__ x,
                           const float* __restrict__ w,
                           const float* __restrict__ msk,
                           float* __restrict__ part,
                           int kchunk) {
    const int wave = threadIdx.x >> 5;
    const int lane = threadIdx.x & 31;
    const int nt   = blockIdx.x * WAVES_PER_BLOCK + wave;
    if (nt >= NTILES) return;                  // whole-wave exit: EXEC stays all-1s
    const int s    = blockIdx.y;               // K-split index
    const int n0   = nt * 16;
    const int half = lane >> 4;                // 0: K pair {0,1}, 1: K pair {2,3}
    const int l15  = lane & 15;
    const int kk0  = s * kchunk + 2 * half;    // first K element this lane touches

    // Per-lane base pointers (A: M = l15 / l15+16 via RSTEP; B: N = n0 + l15).
    // All 8-byte aligned: row strides and kk0 are even in float units.
    const v2f* __restrict__ wp = (const v2f*)(w   + (size_t)(n0 + l15) * IN_DIM + kk0);
    const v2f* __restrict__ xp = (const v2f*)(x   + (size_t)l15        * IN_DIM + kk0);
    const v2f* __restrict__ mp = (const v2f*)(msk + (size_t)l15        * IN_DIM + kk0);

    v8f acc0 = {};   // rows 0..15
    v8f acc1 = {};   // rows 16..31

    const int iters = kchunk >> 3;             // 8 K (2 WMMA k-steps) per iteration
    #pragma unroll 2
    for (int t = 0; t < iters; ++t) {
        // B fragments: weight.T[kk, n] == weight[n, kk]; streamed, non-temporal
        v2f b0 = __builtin_nontemporal_load(wp);
        v2f b1 = __builtin_nontemporal_load(wp + 2);
        // A fragments with mask folded in (L2-resident, regular temporal)
        v2f a00 = xp[0]         * mp[0];
        v2f a10 = xp[RSTEP]     * mp[RSTEP];
        v2f a01 = xp[2]         * mp[2];
        v2f a11 = xp[RSTEP + 2] * mp[RSTEP + 2];

        acc0 = __builtin_amdgcn_wmma_f32_16x16x4_f32(
            false, a00, false, b0, (short)0, acc0, false, false);
        acc1 = __builtin_amdgcn_wmma_f32_16x16x4_f32(
            false, a10, false, b0, (short)0, acc1, false, false);
        acc0 = __builtin_amdgcn_wmma_f32_16x16x4_f32(
            false, a01, false, b1, (short)0, acc0, false, false);
        acc1 = __builtin_amdgcn_wmma_f32_16x16x4_f32(
            false, a11, false, b1, (short)0, acc1, false, false);

        wp += 4; xp += 4; mp += 4;             // advance 8 floats (32 bytes)
    }

    // C/D layout: VGPR v -> M = v + 8*half (lanes 0-15: M=v, 16-31: M=v+8), N = l15
    float* __restrict__ p = part + (size_t)s * BATCH * OUT_DIM;
    #pragma unroll
    for (int v = 0; v < 8; ++v) {
        const int m = v + 8 * half;
        p[(size_t)m        * OUT_DIM + n0 + l15] = acc0[v];
        p[(size_t)(m + 16) * OUT_DIM + n0 + l15] = acc1[v];
    }
}

// Sum the KSPLIT partial products into the final output.
__global__ __launch_bounds__(256)
void reduce_partials(const float* __restrict__ part, float* __restrict__ out, int ks) {
    const int i = blockIdx.x * blockDim.x + threadIdx.x;
    if (i >= BATCH * OUT_DIM) return;
    float acc = 0.0f;
    for (int j = 0; j < ks; ++j)
        acc += part[(size_t)j * BATCH * OUT_DIM + i];
    out[i] = acc;
}

extern "C" void kernel_launch(void* const* d_in, const int* in_sizes, int n_in,
                              void* d_out, int out_size, void* d_ws, size_t ws_size,
                              hipStream_t stream) {
    const float* enc_x  = (const float*)d_in[0];   // [32, 18496]
    const float* weight = (const float*)d_in[1];   // [4624, 18496]
    const float* mask   = (const float*)d_in[2];   // [32, 18496]
    float* out = (float*)d_out;                    // [32, 4624]

    const size_t BO = (size_t)BATCH * OUT_DIM;
    const size_t need = (size_t)KSPLIT * BO * sizeof(float);

    if (ws_size >= need) {
        // K split 8 ways for wave-level parallelism (2312 WMMA waves),
        // partials in workspace, then reduce.
        float* part = (float*)d_ws;
        dim3 grid((NTILES + WAVES_PER_BLOCK - 1) / WAVES_PER_BLOCK, KSPLIT);
        avgpool_toeplitz_wmma<<<grid, WAVES_PER_BLOCK * 32, 0, stream>>>(
            enc_x, weight, mask, part, IN_DIM / KSPLIT);
        reduce_partials<<<(int)((BO + 255) / 256), 256, 0, stream>>>(part, out, KSPLIT);
    } else {
        // Fallback: no K-split, write final results directly (still deterministic).
        dim3 grid((NTILES + WAVES_PER_BLOCK - 1) / WAVES_PER_BLOCK, 1);
        avgpool_toeplitz_wmma<<<grid, WAVES_PER_BLOCK * 32, 0, stream>>>(
            enc_x, weight, mask, out, IN_DIM);
    }
}